// QuantumByzantineDetector_10952166605275
// MI455X (gfx1250) — compile-verified
//
#include <hip/hip_runtime.h>
#include <hip/hip_bf16.h>
#include <math.h>

// ---------------------------------------------------------------------------
// Problem constants
// ---------------------------------------------------------------------------
#define BATCH   1024
#define DIMEN   262144
#define NQ      4
#define NSTATE  16            // 2^NQ
#define NLAYERS 3
#define CHUNK   (DIMEN / NQ)  // 65536 floats per (sample, qubit) chunk

typedef float v2f __attribute__((ext_vector_type(2)));
typedef float v4f __attribute__((ext_vector_type(4)));
typedef float v8f __attribute__((ext_vector_type(8)));

__device__ __forceinline__ float clean_f32(float v) {
    // nan_to_num(nan=0, posinf=0, neginf=0)
    unsigned u = __float_as_uint(v);
    return ((u & 0x7f800000u) == 0x7f800000u) ? 0.0f : v;
}

// ---------------------------------------------------------------------------
// Kernel 1: streaming chunk-mean reduction (the bandwidth-bound 1 GiB pass).
// One block per (sample, qubit) chunk -> deterministic, no atomics.
// 256 threads * 64 iters * v4f = 65536 floats, coalesced b128 NT loads.
// ---------------------------------------------------------------------------
__global__ __launch_bounds__(256)
void qbd_reduce_feats(const float* __restrict__ x, float* __restrict__ feats) {
    const int bq  = blockIdx.x;                 // 0 .. BATCH*NQ-1 (chunks are contiguous)
    const int tid = threadIdx.x;
    const v4f* __restrict__ p = (const v4f*)(x + (long long)bq * CHUNK);

    float s = 0.0f;
    #pragma unroll 4
    for (int it = 0; it < CHUNK / 4 / 256; ++it) {       // 64 iterations
        v4f v = __builtin_nontemporal_load(&p[tid + it * 256]);
        s += clean_f32(v.x) + clean_f32(v.y) + clean_f32(v.z) + clean_f32(v.w);
    }

    __shared__ float red[256];
    red[tid] = s;
    __syncthreads();
    #pragma unroll
    for (int off = 128; off > 0; off >>= 1) {
        if (tid < off) red[tid] += red[tid + off];
        __syncthreads();
    }
    if (tid == 0) feats[bq] = red[0] * (1.0f / (float)CHUNK);
}

// ---------------------------------------------------------------------------
// Kernel 2: build the fixed 16x16 complex unitary U of the 3 entangling
// layers (weights are shared across the batch, so the layer stack is one
// constant linear operator).  One thread per basis column.
// ---------------------------------------------------------------------------
__global__ void qbd_build_U(const float* __restrict__ w,
                            float* __restrict__ Ure, float* __restrict__ Uim) {
    const int t = threadIdx.x;      // basis-state column
    if (t >= NSTATE) return;

    float re[NSTATE], im[NSTATE];
    for (int i = 0; i < NSTATE; ++i) { re[i] = (i == t) ? 1.0f : 0.0f; im[i] = 0.0f; }

    for (int l = 0; l < NLAYERS; ++l) {
        // Rot(phi, theta, omega) = RZ(omega) RY(theta) RZ(phi) on each qubit
        for (int q = 0; q < NQ; ++q) {
            const float phi = w[l * 12 + q * 3 + 0];
            const float th  = w[l * 12 + q * 3 + 1];
            const float om  = w[l * 12 + q * 3 + 2];
            const float c = cosf(0.5f * th), s = sinf(0.5f * th);
            const float apw = 0.5f * (phi + om), ad = 0.5f * (phi - om);
            const float g00r =  cosf(apw) * c, g00i = -sinf(apw) * c;
            const float g11r =  cosf(apw) * c, g11i =  sinf(apw) * c;
            const float g01r = -cosf(ad)  * s, g01i = -sinf(ad)  * s;
            const float g10r =  cosf(ad)  * s, g10i = -sinf(ad)  * s;
            const int bit = 1 << (3 - q);            // qubit q <-> bit (3-q)
            for (int i = 0; i < NSTATE; ++i) {
                if (i & bit) continue;
                const int j = i | bit;
                const float a0r = re[i], a0i = im[i], a1r = re[j], a1i = im[j];
                re[i] = g00r * a0r - g00i * a0i + g01r * a1r - g01i * a1i;
                im[i] = g00r * a0i + g00i * a0r + g01r * a1i + g01i * a1r;
                re[j] = g10r * a0r - g10i * a0i + g11r * a1r - g11i * a1i;
                im[j] = g10r * a0i + g10i * a0r + g11r * a1i + g11i * a1r;
            }
        }
        // CNOT ring, StronglyEntanglingLayers range r = l % (n-1) + 1
        const int r = l % (NQ - 1) + 1;
        for (int q = 0; q < NQ; ++q) {
            const int cbit = 1 << (3 - q);
            const int tbit = 1 << (3 - ((q + r) & (NQ - 1)));
            for (int i = 0; i < NSTATE; ++i) {
                if ((i & cbit) && !(i & tbit)) {
                    const int j = i | tbit;
                    float tr = re[i], ti = im[i];
                    re[i] = re[j]; im[i] = im[j];
                    re[j] = tr;    im[j] = ti;
                }
            }
        }
    }
    for (int i = 0; i < NSTATE; ++i) {
        Ure[i * NSTATE + t] = re[i];
        Uim[i * NSTATE + t] = im[i];
    }
}

// ---------------------------------------------------------------------------
// Kernel 3: final = U * s(b) for 16 samples per wave via V_WMMA_F32_16X16X4_F32.
// Complex GEMM = 4 real WMMA accumulations per K-step (K=16 -> 4 steps).
// A layout: lane L holds A[L&15][v + 2*(L>>4)]    (2 VGPRs)
// B layout: lane L holds B[v + 2*(L>>4)][L&15]    (2 VGPRs)
// D layout: lane L, vgpr r holds D[r + 8*(L>>4)][L&15]
// ---------------------------------------------------------------------------
__global__ __launch_bounds__(32)
void qbd_apply_U(const float* __restrict__ feats,
                 const float* __restrict__ Ure, const float* __restrict__ Uim,
                 float* __restrict__ out) {
    const int lane   = threadIdx.x;      // 0..31, wave32
    const int half   = lane >> 4;
    const int col    = lane & 15;        // sample column in tile == A row m
    const int sample = blockIdx.x * 16 + col;

    // --- encode: clip + per-sample min/max scale to [0, pi] ---
    float f[NQ], fmin = 1e30f, fmax = -1e30f;
    #pragma unroll
    for (int q = 0; q < NQ; ++q) {
        float v = clean_f32(feats[sample * NQ + q]);
        v = fminf(1.0f, fmaxf(-1.0f, v));
        f[q] = v;
        fmin = fminf(fmin, v);
        fmax = fmaxf(fmax, v);
    }
    const float PI = 3.14159265358979323846f;
    const float sc = PI / (fmax - fmin + 1e-9f);
    float cq[NQ], sq[NQ];
    #pragma unroll
    for (int q = 0; q < NQ; ++q) {
        const float ang = (f[q] - fmin) * sc;
        cq[q] = cosf(0.5f * ang);
        sq[q] = sinf(0.5f * ang);
    }

    // --- initial state s(b): amp(i) = (-i)^popcount * prod(cos / sin) ---
    float Sre[NSTATE], Sim[NSTATE];
    #pragma unroll
    for (int i = 0; i < NSTATE; ++i) {
        float mag = 1.0f; int pc = 0;
        #pragma unroll
        for (int q = 0; q < NQ; ++q) {
            if (i & (1 << (3 - q))) { mag *= sq[q]; ++pc; }
            else                    { mag *= cq[q]; }
        }
        switch (pc & 3) {
            case 0: Sre[i] =  mag; Sim[i] =  0.0f; break;
            case 1: Sre[i] =  0.0f; Sim[i] = -mag; break;
            case 2: Sre[i] = -mag; Sim[i] =  0.0f; break;
            default: Sre[i] = 0.0f; Sim[i] =  mag; break;
        }
    }

    // --- complex 16x16 x 16x16 GEMM on the matrix core ---
    v8f dre = {}; v8f dim = {};
    const int m = col;
    #pragma unroll
    for (int kb = 0; kb < NSTATE; kb += 4) {
        const int k0 = kb + 2 * half;
        v2f aRe, aIm, aImN, bRe, bIm;
        aRe.x = Ure[m * NSTATE + k0];     aRe.y = Ure[m * NSTATE + k0 + 1];
        aIm.x = Uim[m * NSTATE + k0];     aIm.y = Uim[m * NSTATE + k0 + 1];
        aImN  = -aIm;
        bRe.x = Sre[k0];  bRe.y = Sre[k0 + 1];
        bIm.x = Sim[k0];  bIm.y = Sim[k0 + 1];
        // Fre += Ure*Sre - Uim*Sim ; Fim += Ure*Sim + Uim*Sre
        dre = __builtin_amdgcn_wmma_f32_16x16x4_f32(false, aRe,  false, bRe, (short)0, dre, false, false);
        dre = __builtin_amdgcn_wmma_f32_16x16x4_f32(false, aImN, false, bIm, (short)0, dre, false, false);
        dim = __builtin_amdgcn_wmma_f32_16x16x4_f32(false, aRe,  false, bIm, (short)0, dim, false, false);
        dim = __builtin_amdgcn_wmma_f32_16x16x4_f32(false, aIm,  false, bRe, (short)0, dim, false, false);
    }

    // --- <Z_3>: sign by parity of output state's LSB (last qubit) ---
    float z = 0.0f;
    #pragma unroll
    for (int r = 0; r < 8; ++r) {
        const int M = r + 8 * half;
        const float p = dre[r] * dre[r] + dim[r] * dim[r];
        z += (M & 1) ? -p : p;
    }
    z += __shfl_xor(z, 16, 32);          // combine the two lane-halves (same N)

    if (lane < 16) {
        const float e = 0.5f * (1.0f + z);
        out[sample] = e * e;
    }
}

// ---------------------------------------------------------------------------
// Launch
// ---------------------------------------------------------------------------
extern "C" void kernel_launch(void* const* d_in, const int* in_sizes, int n_in,
                              void* d_out, int out_size, void* d_ws, size_t ws_size,
                              hipStream_t stream) {
    const float* x = (const float*)d_in[0];     // (1024, 262144) f32
    const float* w = (const float*)d_in[1];     // (3, 4, 3) f32
    float* out = (float*)d_out;                 // (1024, 1) f32

    float* feats = (float*)d_ws;                // BATCH*NQ   = 4096 floats
    float* Ure   = feats + BATCH * NQ;          // 256 floats
    float* Uim   = Ure + NSTATE * NSTATE;       // 256 floats

    qbd_reduce_feats<<<BATCH * NQ, 256, 0, stream>>>(x, feats);
    qbd_build_U<<<1, NSTATE, 0, stream>>>(w, Ure, Uim);
    qbd_apply_U<<<BATCH / 16, 32, 0, stream>>>(feats, Ure, Uim, out);
}